// WSD_76725295775984
// MI455X (gfx1250) — compile-verified
//
#include <hip/hip_runtime.h>
#include <hip/hip_bf16.h>

#define D_DIM 2048
#define BLOCK 256

typedef __attribute__((ext_vector_type(2))) float v2f;
typedef __attribute__((ext_vector_type(8))) float v8f;
typedef __attribute__((ext_vector_type(4))) int   v4i;

typedef __attribute__((address_space(1))) v4i* glb_v4i_p;  // global (AS1) int4*
typedef __attribute__((address_space(3))) v4i* lds_v4i_p;  // LDS (AS3) int4*

// Branchless compare-exchange (data are finite normals from randn; min/max fine).
__device__ __forceinline__ void cas(float* s, unsigned i, unsigned q, bool up) {
    float a = s[i], b = s[q];
    float lo = fminf(a, b), hi = fmaxf(a, b);
    s[i] = up ? lo : hi;
    s[q] = up ? hi : lo;
}

// One block per row: async-load both rows into LDS, bitonic-sort each,
// sum |diff|, deterministic block reduction -> partial[row].
__global__ void __launch_bounds__(BLOCK)
wsd_row_kernel(const float* __restrict__ z2,
               const float* __restrict__ z1,
               float* __restrict__ partial) {
    __shared__ float s1[D_DIM];
    __shared__ float s2[D_DIM];
    const unsigned tid = threadIdx.x;
    const unsigned row = blockIdx.x;
    const float* g1 = z1 + (size_t)row * D_DIM;
    const float* g2 = z2 + (size_t)row * D_DIM;

#if defined(__gfx1250__) && __has_builtin(__builtin_amdgcn_global_load_async_to_lds_b128)
    // CDNA5 async DMA path: 16 B per lane per op, 4 ops/thread total.
    #pragma unroll
    for (unsigned i = tid; i < D_DIM / 4; i += BLOCK) {
        __builtin_amdgcn_global_load_async_to_lds_b128(
            (glb_v4i_p)(g1 + 4u * i), (lds_v4i_p)&s1[4u * i], 0, 0);
        __builtin_amdgcn_global_load_async_to_lds_b128(
            (glb_v4i_p)(g2 + 4u * i), (lds_v4i_p)&s2[4u * i], 0, 0);
    }
  #if __has_builtin(__builtin_amdgcn_s_wait_asynccnt)
    __builtin_amdgcn_s_wait_asynccnt(0);
  #else
    asm volatile("s_wait_asynccnt 0" ::: "memory");
  #endif
#else
    for (unsigned i = tid; i < D_DIM; i += BLOCK) {
        s1[i] = g1[i];
        s2[i] = g2[i];
    }
#endif

    // Bitonic sort both rows (ascending). 66 stages for n=2048.
    for (unsigned k = 2; k <= D_DIM; k <<= 1) {
        for (unsigned j = k >> 1; j > 0; j >>= 1) {
            __syncthreads();
            #pragma unroll
            for (unsigned t = 0; t < (D_DIM / 2) / BLOCK; ++t) {
                unsigned p = tid + t * BLOCK;
                unsigned i = ((p & ~(j - 1u)) << 1) | (p & (j - 1u));
                unsigned q = i | j;
                bool up = ((i & k) == 0u);
                cas(s1, i, q, up);
                cas(s2, i, q, up);
            }
        }
    }
    __syncthreads();

    // Per-thread strided |diff| sum (thread X touches only indices == X mod 256).
    float acc = 0.0f;
    for (unsigned i = tid; i < D_DIM; i += BLOCK)
        acc += fabsf(s1[i] - s2[i]);

    s1[tid] = acc;
    __syncthreads();
    // Deterministic tree reduction.
    for (unsigned s = BLOCK / 2; s > 0; s >>= 1) {
        if (tid < s) s1[tid] += s1[tid + s];
        __syncthreads();
    }
    if (tid == 0) partial[row] = s1[0];
}

// Single-wave final reduction. 32 lane partials are summed exactly with one
// v_wmma_f32_16x16x4_f32: A layout (ISA 7.12.2) puts lanes 0-15 at K=0 and
// lanes 16-31 at K=2; with B = all-ones, D[m][n] = partial[m] + partial[m+16]
// for every column n, so sum of the 8 C/D VGPRs on any lane n<16 plus the same
// on lane n+16 is the grand total.
__global__ void __launch_bounds__(32)
wsd_final_kernel(const float* __restrict__ partial, float* __restrict__ out,
                 int nblocks, float inv_count) {
    const unsigned lane = threadIdx.x;  // exactly one wave32, EXEC all ones
    float s = 0.0f;
    for (int i = (int)lane; i < nblocks; i += 32)  // fixed order -> deterministic
        s += partial[i];

#if defined(__gfx1250__) && __has_builtin(__builtin_amdgcn_wmma_f32_16x16x4_f32)
    v2f a; a[0] = s;    a[1] = 0.0f;
    v2f b; b[0] = 1.0f; b[1] = 1.0f;
    v8f c = {};
    c = __builtin_amdgcn_wmma_f32_16x16x4_f32(
            /*neg_a=*/false, a, /*neg_b=*/false, b,
            /*c_mod=*/(short)0, c, /*reuse_a=*/false, /*reuse_b=*/false);
    float t = ((c[0] + c[1]) + (c[2] + c[3])) + ((c[4] + c[5]) + (c[6] + c[7]));
    float tot = t + __shfl_xor(t, 16, 32);  // lane half-sums -> grand total
#else
    float tot = s;
    for (int off = 16; off > 0; off >>= 1)
        tot += __shfl_xor(tot, off, 32);
#endif

    if (lane == 0) out[0] = 1.0f - tot * inv_count;
}

extern "C" void kernel_launch(void* const* d_in, const int* in_sizes, int n_in,
                              void* d_out, int out_size, void* d_ws, size_t ws_size,
                              hipStream_t stream) {
    const float* z2 = (const float*)d_in[0];
    const float* z1 = (const float*)d_in[1];
    const int total = in_sizes[0];
    const int L = total / D_DIM;          // 16384
    float* partial = (float*)d_ws;        // L floats = 64 KB scratch

    wsd_row_kernel<<<dim3((unsigned)L), dim3(BLOCK), 0, stream>>>(z2, z1, partial);

    const float inv_count = 1.0f / ((float)L * (float)D_DIM);  // 2^-25, exact
    wsd_final_kernel<<<dim3(1), dim3(32), 0, stream>>>(partial, (float*)d_out,
                                                       L, inv_count);
}